// IndexedGrid_46256797777984
// MI455X (gfx1250) — compile-verified
//
#include <hip/hip_runtime.h>

typedef float v4f __attribute__((ext_vector_type(4)));
typedef float v2f __attribute__((ext_vector_type(2)));

#define GRID_H 1024
#define GRID_W 1024
#define NPIX   (GRID_H * GRID_W)
#define NBR    17                 // 1 + 2*NUM_FREQ branches per pixel
#define TOTAL  (NPIX * NBR)       // 17,825,792 threads, exactly 69632 * 256

// One thread per (pixel p, branch b) with t = p*17 + b.
// Output floats for thread t are exactly out[8t .. 8t+7]  ->  perfectly
// coalesced 32B-aligned stores across the wave (wave32: 32 lanes * 32B = 1KB/clause).
__global__ __launch_bounds__(256) void indexed_grid_kernel(
    const float* __restrict__ c,
    const float* __restrict__ g0, const float* __restrict__ g1,
    const float* __restrict__ g2, const float* __restrict__ g3,
    const float* __restrict__ g4, const float* __restrict__ g5,
    const float* __restrict__ g6, const float* __restrict__ g7,
    const float* __restrict__ g8,
    float* __restrict__ out)
{
    // Small LDS tables: 9 grid pointers, 9 sizes, 8 frequency factors.
    // Divergent-indexed via ds_load -> avoids per-thread scratch arrays.
    __shared__ const float* sg[9];
    __shared__ float ssz1[9];     // (S - 1) as float
    __shared__ int   ssz[9];      // S
    __shared__ float sfreq[8];    // pi * 1.5^i, i = 1..8

    const int tid = threadIdx.x;
    if (tid == 0) {
        sg[0] = g0; sg[1] = g1; sg[2] = g2; sg[3] = g3; sg[4] = g4;
        sg[5] = g5; sg[6] = g6; sg[7] = g7; sg[8] = g8;
        const int sz[9] = {51, 74, 108, 157, 228, 332, 484, 704, 1024};
#pragma unroll
        for (int i = 0; i < 9; ++i) { ssz[i] = sz[i]; ssz1[i] = (float)(sz[i] - 1); }
        sfreq[0] = 4.71238898038469f;      // pi*1.5
        sfreq[1] = 7.068583470577035f;     // pi*1.5^2
        sfreq[2] = 10.602875205865551f;
        sfreq[3] = 15.904312808798327f;
        sfreq[4] = 23.85646921319749f;
        sfreq[5] = 35.784703819796236f;
        sfreq[6] = 53.677055729694354f;
        sfreq[7] = 80.51558359454153f;     // pi*1.5^8
    }
    __syncthreads();

    const int t = blockIdx.x * 256 + tid;
    if (t >= TOTAL) return;

    const int p = t / NBR;            // pixel index (row-major over 1024x1024)
    const int b = t - p * NBR;        // branch: 0 = raw; odd = sin; even = cos

    // Coords reused by 17 consecutive threads -> served from L0/L2.
    const v2f cc = *(const v2f*)(c + 2 * (size_t)p);

    float u, v;
    int gi;
    if (b == 0) {
        u = cc.x; v = cc.y; gi = 0;
    } else {
        gi = (b + 1) >> 1;                       // frequency / grid index 1..8
        const float f  = sfreq[gi - 1];
        const float a0 = cc.x * f;
        const float a1 = cc.y * f;
        float s0, s1;
        if (b & 1) { s0 = __sinf(a0); s1 = __sinf(a1); }   // v_sin_f32
        else       { s0 = __cosf(a0); s1 = __cosf(a1); }   // v_cos_f32
        u = __builtin_fabsf(s0);
        v = __builtin_fabsf(s1);
    }

    const int          S  = ssz[gi];
    const float        sm = ssz1[gi];
    const float* __restrict__ g = sg[gi];

    // Nearest-lower gather: trunc == floor for u,v in [0,1]; u==1 -> S-1 (in range).
    const int hi = (int)(u * sm);
    const int wi = (int)(v * sm);

    // 8 contiguous f32 per gather, 32B aligned (F = 8) -> two global_load_b128.
    const v4f* __restrict__ src = (const v4f*)(g + ((size_t)(hi * S + wi) << 3));
    const v4f r0 = src[0];
    const v4f r1 = src[1];

    // Output is 544 MiB streamed write-once: non-temporal stores (TH=NT) keep the
    // 192MB L2 free to hold the 64MB of grid tables for the random gathers.
    v4f* __restrict__ dst = (v4f*)(out + ((size_t)t << 3));
    __builtin_nontemporal_store(r0, dst);
    __builtin_nontemporal_store(r1, dst + 1);
}

extern "C" void kernel_launch(void* const* d_in, const int* in_sizes, int n_in,
                              void* d_out, int out_size, void* d_ws, size_t ws_size,
                              hipStream_t stream) {
    (void)in_sizes; (void)n_in; (void)out_size; (void)d_ws; (void)ws_size;

    const float* c  = (const float*)d_in[0];
    const float* g0 = (const float*)d_in[1];
    const float* g1 = (const float*)d_in[2];
    const float* g2 = (const float*)d_in[3];
    const float* g3 = (const float*)d_in[4];
    const float* g4 = (const float*)d_in[5];
    const float* g5 = (const float*)d_in[6];
    const float* g6 = (const float*)d_in[7];
    const float* g7 = (const float*)d_in[8];
    const float* g8 = (const float*)d_in[9];
    float* out = (float*)d_out;

    const int blocks = (TOTAL + 255) / 256;   // 69632, exact
    indexed_grid_kernel<<<dim3(blocks), dim3(256), 0, stream>>>(
        c, g0, g1, g2, g3, g4, g5, g6, g7, g8, out);
}